// GCNModel_4733053960250
// MI455X (gfx1250) — compile-verified
//
#include <hip/hip_runtime.h>
#include <hip/hip_bf16.h>

#define DFEAT 128
#define NNODES 50000

typedef float v2f __attribute__((ext_vector_type(2)));
typedef float v8f __attribute__((ext_vector_type(8)));

// ---------------------------------------------------------------------------
// zero fill
// ---------------------------------------------------------------------------
__global__ void zero_kernel(float* __restrict__ p, int n) {
    int i = blockIdx.x * blockDim.x + threadIdx.x;
    int stride = gridDim.x * blockDim.x;
    for (; i < n; i += stride) p[i] = 0.0f;
}

// ---------------------------------------------------------------------------
// degree counting: deg_out[src[e]] += 1, deg_in[dst[e]] += 1
// ---------------------------------------------------------------------------
__global__ void degree_kernel(const int* __restrict__ src, const int* __restrict__ dst,
                              float* __restrict__ deg_out, float* __restrict__ deg_in, int E) {
    int i = blockIdx.x * blockDim.x + threadIdx.x;
    if (i < E) {
        atomicAdd(&deg_out[src[i]], 1.0f);
        atomicAdd(&deg_in[dst[i]], 1.0f);
    }
}

// norm = clip(deg,1)^-0.5
__global__ void norm_kernel(const float* __restrict__ deg, float* __restrict__ norm, int n) {
    int i = blockIdx.x * blockDim.x + threadIdx.x;
    if (i < n) norm[i] = rsqrtf(fmaxf(deg[i], 1.0f));
}

// ---------------------------------------------------------------------------
// W (128x128 row-major, W[k][j]) -> Wt[j][k] so B-fragments are contiguous
// ---------------------------------------------------------------------------
__global__ void transpose128_kernel(const float* __restrict__ W, float* __restrict__ Wt) {
    int i = blockIdx.x * blockDim.x + threadIdx.x;
    if (i < DFEAT * DFEAT) {
        int k = i >> 7;
        int j = i & 127;
        Wt[j * DFEAT + k] = W[i];
    }
}

// ---------------------------------------------------------------------------
// out[n][:] = (in[n][:] * scale[n]) @ W   via V_WMMA_F32_16X16X4_F32
// block = 256 threads (8 waves); block handles 16 rows; wave w handles
// columns [16w, 16w+16). K=128 marched in steps of 4.
// A-frag lane layout: M = lane%16, K = {kb, kb+1}, kb = kk + 2*(lane/16)
// B-frag lane layout: N = lane%16, K = {kb, kb+1}  (from Wt, contiguous)
// D layout: lane holds col N=lane%16, VGPR i holds row M = i + 8*(lane/16)
// ---------------------------------------------------------------------------
__global__ __launch_bounds__(256) void gemm_scale_wmma_kernel(
    const float* __restrict__ in, const float* __restrict__ scale,
    const float* __restrict__ Wt, float* __restrict__ out) {
    __shared__ float As[16][DFEAT];  // scaled A tile, 8KB

    const int tid  = threadIdx.x;
    const int wave = tid >> 5;
    const int lane = tid & 31;
    const int row0 = blockIdx.x * 16;

    // stage scaled A tile: 2048 elements, 8 per thread, coalesced
    for (int i = tid; i < 16 * DFEAT; i += 256) {
        int r = i >> 7;
        int c = i & 127;
        As[r][c] = in[(size_t)(row0 + r) * DFEAT + c] * scale[row0 + r];
    }
    __syncthreads();

    const int m      = lane & 15;
    const int ncol   = (wave << 4) + (lane & 15);
    const int khalf  = (lane >> 4) << 1;  // 0 or 2

    v8f acc = {};
#pragma unroll
    for (int kk = 0; kk < DFEAT; kk += 4) {
        const int kb = kk + khalf;
        v2f a = *(const v2f*)&As[m][kb];                       // ds_load_b64
        v2f b = *(const v2f*)&Wt[(size_t)ncol * DFEAT + kb];   // global_load_b64
        acc = __builtin_amdgcn_wmma_f32_16x16x4_f32(
            /*neg_a=*/false, a, /*neg_b=*/false, b,
            /*c_mod=*/(short)0, acc, /*reuse_a=*/false, /*reuse_b=*/false);
    }

    const int mrow = (lane >> 4) << 3;  // 0 or 8
#pragma unroll
    for (int i = 0; i < 8; ++i) {
        out[(size_t)(row0 + mrow + i) * DFEAT + ncol] = acc[i];
    }
}

// ---------------------------------------------------------------------------
// agg[dst[e]][j] += proj[src[e]][j]  — one 128-thread block per edge
// ---------------------------------------------------------------------------
__global__ __launch_bounds__(128) void scatter_add_kernel(
    const float* __restrict__ proj, const int* __restrict__ src,
    const int* __restrict__ dst, float* __restrict__ agg, int E) {
    int e = blockIdx.x;
    if (e >= E) return;
    int j = threadIdx.x;
    int s = src[e];
    int d = dst[e];
    atomicAdd(&agg[(size_t)d * DFEAT + j], proj[(size_t)s * DFEAT + j]);
}

// ---------------------------------------------------------------------------
// h = relu(agg * norm_dst[node] + b[j])
// ---------------------------------------------------------------------------
__global__ void finalize_kernel(const float* __restrict__ agg,
                                const float* __restrict__ norm_dst,
                                const float* __restrict__ b,
                                float* __restrict__ h, int n) {
    int i = blockIdx.x * blockDim.x + threadIdx.x;
    if (i < n) {
        int node = i >> 7;
        int j    = i & 127;
        float v  = agg[i] * norm_dst[node] + b[j];
        h[i] = fmaxf(v, 0.0f);
    }
}

// ---------------------------------------------------------------------------
// per-edge dot product: one wave32 per edge, float4 per lane, shfl reduce
// ---------------------------------------------------------------------------
__global__ __launch_bounds__(256) void edge_dot_kernel(
    const float* __restrict__ h, const int* __restrict__ u,
    const int* __restrict__ v, float* __restrict__ out, int E) {
    int widx = blockIdx.x * (blockDim.x >> 5) + (threadIdx.x >> 5);
    int lane = threadIdx.x & 31;
    if (widx >= E) return;
    const float4* a = (const float4*)&h[(size_t)u[widx] * DFEAT];
    const float4* b = (const float4*)&h[(size_t)v[widx] * DFEAT];
    float4 x = a[lane];
    float4 y = b[lane];
    float s = x.x * y.x + x.y * y.y + x.z * y.z + x.w * y.w;
#pragma unroll
    for (int off = 16; off > 0; off >>= 1) s += __shfl_xor(s, off, 32);
    if (lane == 0) out[widx] = s;
}

// ---------------------------------------------------------------------------
// launch
// ---------------------------------------------------------------------------
extern "C" void kernel_launch(void* const* d_in, const int* in_sizes, int n_in,
                              void* d_out, int out_size, void* d_ws, size_t ws_size,
                              hipStream_t stream) {
    const float* x       = (const float*)d_in[0];
    const int*   src     = (const int*)d_in[1];
    const int*   dst     = (const int*)d_in[2];
    const int*   pos_src = (const int*)d_in[3];
    const int*   pos_dst = (const int*)d_in[4];
    const int*   neg_src = (const int*)d_in[5];
    const int*   neg_dst = (const int*)d_in[6];
    const float* W1      = (const float*)d_in[7];
    const float* b1      = (const float*)d_in[8];
    const float* W2      = (const float*)d_in[9];
    const float* b2      = (const float*)d_in[10];

    const int N  = in_sizes[0] / DFEAT;  // 50000
    const int E  = in_sizes[1];          // 800000
    const int Ep = in_sizes[3];          // 200000
    const int En = in_sizes[5];          // 1000000

    float* ws = (float*)d_ws;
    float* deg_out  = ws;                 ws += N;
    float* deg_in   = ws;                 ws += N;
    float* norm_src = ws;                 ws += N;
    float* norm_dst = ws;                 ws += N;
    float* Wt       = ws;                 ws += DFEAT * DFEAT;
    float* proj     = ws;                 ws += (size_t)N * DFEAT;
    float* agg      = ws;                 ws += (size_t)N * DFEAT;
    float* h        = ws;                 ws += (size_t)N * DFEAT;

    float* out_pos = (float*)d_out;
    float* out_neg = out_pos + Ep;

    const int NF = N * DFEAT;

    // --- degrees + norms ---
    zero_kernel<<<1024, 256, 0, stream>>>(deg_out, 2 * N);  // deg_out,deg_in contiguous
    degree_kernel<<<(E + 255) / 256, 256, 0, stream>>>(src, dst, deg_out, deg_in, E);
    norm_kernel<<<(N + 255) / 256, 256, 0, stream>>>(deg_out, norm_src, N);
    norm_kernel<<<(N + 255) / 256, 256, 0, stream>>>(deg_in, norm_dst, N);

    // --- layer 1 ---
    transpose128_kernel<<<(DFEAT * DFEAT + 255) / 256, 256, 0, stream>>>(W1, Wt);
    gemm_scale_wmma_kernel<<<N / 16, 256, 0, stream>>>(x, norm_src, Wt, proj);
    zero_kernel<<<4096, 256, 0, stream>>>(agg, NF);
    scatter_add_kernel<<<E, 128, 0, stream>>>(proj, src, dst, agg, E);
    finalize_kernel<<<(NF + 255) / 256, 256, 0, stream>>>(agg, norm_dst, b1, h, NF);

    // --- layer 2 ---
    transpose128_kernel<<<(DFEAT * DFEAT + 255) / 256, 256, 0, stream>>>(W2, Wt);
    gemm_scale_wmma_kernel<<<N / 16, 256, 0, stream>>>(h, norm_src, Wt, proj);
    zero_kernel<<<4096, 256, 0, stream>>>(agg, NF);
    scatter_add_kernel<<<E, 128, 0, stream>>>(proj, src, dst, agg, E);
    finalize_kernel<<<(NF + 255) / 256, 256, 0, stream>>>(agg, norm_dst, b2, h, NF);

    // --- edge scores (pos then neg, concatenated in d_out) ---
    edge_dot_kernel<<<(Ep + 7) / 8, 256, 0, stream>>>(h, pos_src, pos_dst, out_pos, Ep);
    edge_dot_kernel<<<(En + 7) / 8, 256, 0, stream>>>(h, neg_src, neg_dst, out_neg, En);
}